// GNN_bench_12421045420924
// MI455X (gfx1250) — compile-verified
//
#include <hip/hip_runtime.h>
#include <hip/hip_bf16.h>

// ---------------------------------------------------------------------------
// GNN bench for MI455X (gfx1250, wave32, WMMA + TDM).
// fp32 end-to-end: workload is bandwidth/atomic bound (~1 GB effective traffic
// at 23.3 TB/s), GEMM FLOPs (~21 GF) are negligible, so V_WMMA_F32_16X16X4_F32
// keeps full reference precision on the matrix ops. Weights are staged into
// LDS with the Tensor Data Mover (tensor_load_to_lds + s_wait_tensorcnt).
// ---------------------------------------------------------------------------

typedef float v2f __attribute__((ext_vector_type(2)));
typedef float v8f __attribute__((ext_vector_type(8)));
typedef unsigned int u32x4 __attribute__((ext_vector_type(4)));
typedef int i32x4 __attribute__((ext_vector_type(4)));
typedef int i32x8 __attribute__((ext_vector_type(8)));

#define N_NODES 50000
#define N_EDGES 800000
#define N_GRAPH 500
#define NH      128
#define EFEAT   16
#define NLAYER  4
#define NCLASS  10

#if defined(__HIP_DEVICE_COMPILE__) && __has_builtin(__builtin_amdgcn_tensor_load_to_lds)
#define HAVE_TDM 1
#else
#define HAVE_TDM 0
#endif

__device__ __forceinline__ v8f wmma_f32_16x16x4(v2f a, v2f b, v8f c) {
  // 8 args: (neg_a, A, neg_b, B, c_mod, C, reuse_a, reuse_b)
  return __builtin_amdgcn_wmma_f32_16x16x4_f32(false, a, false, b, (short)0, c,
                                               false, false);
}

#if HAVE_TDM
// Issue a TDM 2D tile load (f32, tile == tensor) into LDS. Wave-level op:
// call from one wave only; EXEC is ignored by TENSOR_LOAD_TO_LDS.
__device__ __forceinline__ void tdm_load_2d_f32(const float* gptr,
                                                float* lds_dst, int dim0,
                                                int dim1, int stride0) {
  const unsigned long long ga = (unsigned long long)(size_t)gptr;
  const unsigned lds_off = (unsigned)(size_t)(void*)lds_dst;  // aperture low32
  u32x4 g0;
  g0[0] = 1u;                                   // count=1, user descriptor
  g0[1] = lds_off;                              // lds_addr (bytes)
  g0[2] = (unsigned)(ga & 0xffffffffu);         // global_addr[31:0]
  g0[3] = (unsigned)((ga >> 32) & 0x01ffffffu)  // global_addr[56:32]
          | (2u << 30);                         // type = 2 ("image")
  i32x8 g1;
  g1[0] = 0x00020000;                            // data_size = 4B, mask=0
  g1[1] = (dim0 & 0xffff) << 16;                 // tensor_dim0[15:0]
  g1[2] = ((dim0 >> 16) & 0xffff) |              // tensor_dim0[31:16]
          ((dim1 & 0xffff) << 16);               // tensor_dim1[15:0]
  g1[3] = ((dim1 >> 16) & 0xffff) |              // tensor_dim1[31:16]
          ((dim0 & 0xffff) << 16);               // tile_dim0 = dim0
  g1[4] = dim1 & 0xffff;                         // tile_dim1 = dim1, dim2 = 0
  g1[5] = stride0;                               // tensor_dim0_stride[31:0]
  g1[6] = 0;
  g1[7] = 0;
  const i32x4 z4 = {0, 0, 0, 0};
#if __clang_major__ >= 23
  const i32x8 z8 = {0, 0, 0, 0, 0, 0, 0, 0};
  __builtin_amdgcn_tensor_load_to_lds(g0, g1, z4, z4, z8, 0);
#else
  __builtin_amdgcn_tensor_load_to_lds(g0, g1, z4, z4, 0);
#endif
}
#endif  // HAVE_TDM

// ---------------------------------------------------------------------------
// C[M x 128] = (A1 (+ A2)) @ W[128 x 128] + bias
// Block = 256 thr = 8 waves; each wave -> 16 rows x 128 cols (8 v8f accums).
// W is TDM-DMA'd into LDS staging, then repacked into WMMA B-fragment order:
//   frag float index = ((ks*8 + j)*32 + lane)*2 + comp
//   where k = 4*ks + 2*(lane>>4) + comp, col = 16*j + (lane&15)
// so the inner loop is a single aligned ds_load_b64 per fragment.
// Dynamic LDS: [0,16384) = fragments, [16384,32768) = TDM staging (128 KB).
// ---------------------------------------------------------------------------
template <bool HAS_A2>
__global__ __launch_bounds__(256) void gemm128_bias(
    const float* __restrict__ A1, const float* __restrict__ A2,
    const float* __restrict__ W, const float* __restrict__ bias,
    float* __restrict__ C, int M) {
  extern __shared__ float lds[];
  float* bfrag = lds;  // 16384 floats = 64 KB

  const int tid = threadIdx.x;
#if HAVE_TDM
  float* wstage = lds + 16384;  // 64 KB staging, row-major W
  if (tid < 32) {  // wave 0 issues the DMA and waits on TENSORcnt
    tdm_load_2d_f32(W, wstage, NH, NH, NH);
    __builtin_amdgcn_s_wait_tensorcnt(0);
  }
  __syncthreads();
#endif
  for (int idx = tid; idx < 32 * 8 * 32 * 2; idx += 256) {
    const int comp = idx & 1;
    const int ln   = (idx >> 1) & 31;
    const int j    = (idx >> 6) & 7;
    const int ks   = idx >> 9;
    const int k    = 4 * ks + 2 * (ln >> 4) + comp;
    const int col  = 16 * j + (ln & 15);
#if HAVE_TDM
    bfrag[idx] = wstage[k * NH + col];
#else
    bfrag[idx] = W[k * NH + col];
#endif
  }
  __syncthreads();

  const int lane = tid & 31;
  const int wv   = tid >> 5;
  const int half = lane >> 4;
  const int lrow = lane & 15;
  const long r0  = (long)blockIdx.x * 128 + (long)wv * 16;
  if (r0 + 16 > (long)M) return;  // wave-uniform; EXEC stays all-ones

  v8f acc[8];
#pragma unroll
  for (int j = 0; j < 8; ++j)
    acc[j] = (v8f){0.f, 0.f, 0.f, 0.f, 0.f, 0.f, 0.f, 0.f};

  const float* arow1 = A1 + (r0 + lrow) * NH + 2 * half;
  const float* arow2 = HAS_A2 ? (A2 + (r0 + lrow) * NH + 2 * half) : nullptr;

  for (int ks = 0; ks < 32; ++ks) {
    v2f a = *(const v2f*)(arow1 + 4 * ks);
    if (HAS_A2) {  // compile-time: branch-free inner loop in both variants
      v2f a2 = *(const v2f*)(arow2 + 4 * ks);
      a += a2;
    }
    const v2f* bf = (const v2f*)&bfrag[ks * 512];
#pragma unroll
    for (int j = 0; j < 8; ++j) {
      v2f b = bf[j * 32 + lane];
      acc[j] = wmma_f32_16x16x4(a, b, acc[j]);
    }
  }

#pragma unroll
  for (int j = 0; j < 8; ++j) {
    const int col = 16 * j + lrow;
    const float bv = bias[col];
#pragma unroll
    for (int r = 0; r < 8; ++r) {
      C[(r0 + 8 * half + r) * NH + col] = acc[j][r] + bv;
    }
  }
}

// ---------------------------------------------------------------------------
// Fused edge stage: msg = edge_attr @ We + be + h[src]; agg[dst] += msg
// (scatter via global_atomic_add_f32; agg is L2-resident, 25.6 MB).
// Each wave handles 16 edges: K=16 -> 4 WMMA k-steps x 8 col tiles.
// ---------------------------------------------------------------------------
__global__ __launch_bounds__(256) void edge_conv(
    const float* __restrict__ h, const float* __restrict__ ea,
    const int* __restrict__ src, const int* __restrict__ dst,
    const float* __restrict__ We, const float* __restrict__ be,
    float* __restrict__ agg) {
  __shared__ float bfrag[4 * 8 * 32 * 2];  // 2048 floats = 8 KB
  __shared__ float sbias[NH];

  const int tid = threadIdx.x;
  for (int idx = tid; idx < 4 * 8 * 32 * 2; idx += 256) {
    const int comp = idx & 1;
    const int ln   = (idx >> 1) & 31;
    const int j    = (idx >> 6) & 7;
    const int ks   = idx >> 9;
    const int k    = 4 * ks + 2 * (ln >> 4) + comp;
    const int col  = 16 * j + (ln & 15);
    bfrag[idx] = We[k * NH + col];
  }
  if (tid < NH) sbias[tid] = be[tid];
  __syncthreads();

  const int lane = tid & 31;
  const int wv   = tid >> 5;
  const int half = lane >> 4;
  const int lrow = lane & 15;
  const long e0  = ((long)blockIdx.x * 8 + wv) * 16;

  v8f acc[8];
#pragma unroll
  for (int j = 0; j < 8; ++j)
    acc[j] = (v8f){0.f, 0.f, 0.f, 0.f, 0.f, 0.f, 0.f, 0.f};

  const float* arow = ea + (e0 + lrow) * EFEAT + 2 * half;
#pragma unroll
  for (int ks = 0; ks < 4; ++ks) {
    v2f a = *(const v2f*)(arow + 4 * ks);
    const v2f* bf = (const v2f*)&bfrag[ks * 512];
#pragma unroll
    for (int j = 0; j < 8; ++j) {
      v2f b = bf[j * 32 + lane];
      acc[j] = wmma_f32_16x16x4(a, b, acc[j]);
    }
  }

#pragma unroll
  for (int r = 0; r < 8; ++r) {
    const long e = e0 + 8 * half + r;
    const int  s = src[e];
    const int  d = dst[e];
    const float* hs = h + (long)s * NH;
    float*       ag = agg + (long)d * NH;
#pragma unroll
    for (int j = 0; j < 8; ++j) {
      const int c = 16 * j + lrow;
      atomicAdd(&ag[c], acc[j][r] + sbias[c] + hs[c]);
    }
  }
}

// ---------------------------------------------------------------------------
// BN column statistics: stats[0..127] = sum, stats[128..255] = sum of squares.
// Block handles 256 rows; LDS pairwise reduce then f32 atomics.
// ---------------------------------------------------------------------------
__global__ __launch_bounds__(256) void bn_stats(const float* __restrict__ z,
                                                float* __restrict__ stats,
                                                int M) {
  __shared__ float red[512];
  const int tid = threadIdx.x;
  const int c   = tid & 127;
  const int rh  = tid >> 7;  // 0 or 1
  const long r0 = (long)blockIdx.x * 256;
  float s1 = 0.f, s2 = 0.f;
  for (int i = rh; i < 256; i += 2) {
    const long r = r0 + i;
    if (r < (long)M) {
      const float x = z[r * NH + c];
      s1 += x;
      s2 += x * x;
    }
  }
  red[tid]       = s1;
  red[256 + tid] = s2;
  __syncthreads();
  if (tid < 128) {
    atomicAdd(&stats[c],       red[tid]       + red[tid + 128]);
    atomicAdd(&stats[128 + c], red[256 + tid] + red[256 + tid + 128]);
  }
}

// ---------------------------------------------------------------------------
// h_next = relu((z - mu) * rsqrt(var + eps) * gamma + beta) + h_prev
// float4-vectorized (128 cols / 4).
// ---------------------------------------------------------------------------
__global__ __launch_bounds__(256) void bn_apply(
    const float* __restrict__ z, const float* __restrict__ stats,
    const float* __restrict__ gamma, const float* __restrict__ beta,
    const float* __restrict__ hprev, float* __restrict__ hnext, int M) {
  const long idx = (long)blockIdx.x * 256 + threadIdx.x;
  const long total = (long)M * (NH / 4);
  if (idx >= total) return;
  const long e = idx * 4;
  const int  c = (int)(e & (NH - 1));
  const float invN = 1.f / (float)M;
  const float4 z4 = *(const float4*)(z + e);
  const float4 h4 = *(const float4*)(hprev + e);
  float zi[4] = {z4.x, z4.y, z4.z, z4.w};
  float hi[4] = {h4.x, h4.y, h4.z, h4.w};
  float oi[4];
#pragma unroll
  for (int i = 0; i < 4; ++i) {
    const float mu  = stats[c + i] * invN;
    const float var = stats[128 + c + i] * invN - mu * mu;
    float v = (zi[i] - mu) * __frsqrt_rn(var + 1e-5f) * gamma[c + i] +
              beta[c + i];
    oi[i] = fmaxf(v, 0.f) + hi[i];
  }
  *(float4*)(hnext + e) = make_float4(oi[0], oi[1], oi[2], oi[3]);
}

// ---------------------------------------------------------------------------
// Global add-pool: pooled[batch[row]][c] += h[row][c]  (L2-resident target)
// ---------------------------------------------------------------------------
__global__ __launch_bounds__(256) void pool_add(const float* __restrict__ h,
                                                const int* __restrict__ batch,
                                                float* __restrict__ pooled) {
  const long idx = (long)blockIdx.x * 256 + threadIdx.x;
  if (idx >= (long)N_NODES * NH) return;
  const int row = (int)(idx >> 7);
  const int c   = (int)(idx & (NH - 1));
  const int g   = batch[row];
  atomicAdd(&pooled[(long)g * NH + c], h[idx]);
}

// ---------------------------------------------------------------------------
// JK head: out[g][cls] += bias[cls] + pooled[g][:] . W[:, cls]   (tiny GEMM)
// ---------------------------------------------------------------------------
__global__ __launch_bounds__(256) void jk_head(const float* __restrict__ pooled,
                                               const float* __restrict__ W,
                                               const float* __restrict__ b,
                                               float* __restrict__ out) {
  const int idx = blockIdx.x * 256 + threadIdx.x;
  if (idx >= N_GRAPH * NCLASS) return;
  const int g = idx / NCLASS;
  const int c = idx % NCLASS;
  const float* p = pooled + (long)g * NH;
  float acc = b[c];
#pragma unroll 8
  for (int k = 0; k < NH; ++k) acc += p[k] * W[k * NCLASS + c];
  out[idx] += acc;
}

// ---------------------------------------------------------------------------
extern "C" void kernel_launch(void* const* d_in, const int* in_sizes, int n_in,
                              void* d_out, int out_size, void* d_ws,
                              size_t ws_size, hipStream_t stream) {
  (void)in_sizes; (void)n_in; (void)out_size; (void)ws_size;

  const float* h_in      = (const float*)d_in[0];
  const float* edge_attr = (const float*)d_in[1];
  const int*   ei        = (const int*)d_in[2];
  const int*   batch     = (const int*)d_in[5];
  const float* emb_W     = (const float*)d_in[6];
  const float* emb_b     = (const float*)d_in[7];
  const float* conv_W    = (const float*)d_in[8];
  const float* conv_b    = (const float*)d_in[9];
  const float* conv_We   = (const float*)d_in[10];
  const float* conv_be   = (const float*)d_in[11];
  const float* bn_g      = (const float*)d_in[12];
  const float* bn_b      = (const float*)d_in[13];
  const float* jk_W      = (const float*)d_in[14];
  const float* jk_b      = (const float*)d_in[15];
  float* out = (float*)d_out;

  const int* srcp = ei;
  const int* dstp = ei + N_EDGES;

  // Workspace layout (floats)
  const size_t HN = (size_t)N_NODES * NH;  // 6.4 M floats
  float* ws     = (float*)d_ws;
  float* hbuf   = ws;             // 5 * HN   (h_list[0..4])
  float* agg    = ws + 5 * HN;    // HN
  float* zbuf   = agg + HN;       // HN
  float* stats  = zbuf + HN;      // 256
  float* pooled = stats + 256;    // N_GRAPH * NH

  const int gemm_grid  = (N_NODES + 127) / 128;  // 391
  const int edge_grid  = N_EDGES / 128;          // 6250
  const int stat_grid  = (N_NODES + 255) / 256;  // 196
  const int bnap_grid  = (N_NODES * (NH / 4) + 255) / 256;
  const int pool_grid  = (int)((HN + 255) / 256);
  const int head_grid  = (N_GRAPH * NCLASS + 255) / 256;
  const size_t gemm_lds = 2u * 16384u * sizeof(float);  // frag + TDM staging

  // Embedding: h0 = h_in @ emb_W + emb_b
  gemm128_bias<false><<<gemm_grid, 256, gemm_lds, stream>>>(
      h_in, nullptr, emb_W, emb_b, hbuf, N_NODES);

  for (int l = 0; l < NLAYER; ++l) {
    const float* hl = hbuf + (size_t)l * HN;
    float* hn       = hbuf + (size_t)(l + 1) * HN;
    hipMemsetAsync(agg, 0, HN * sizeof(float), stream);
    hipMemsetAsync(stats, 0, 256 * sizeof(float), stream);

    edge_conv<<<edge_grid, 256, 0, stream>>>(
        hl, edge_attr, srcp, dstp, conv_We + (size_t)l * EFEAT * NH,
        conv_be + (size_t)l * NH, agg);

    gemm128_bias<true><<<gemm_grid, 256, gemm_lds, stream>>>(
        hl, agg, conv_W + (size_t)l * NH * NH, conv_b + (size_t)l * NH, zbuf,
        N_NODES);

    bn_stats<<<stat_grid, 256, 0, stream>>>(zbuf, stats, N_NODES);
    bn_apply<<<bnap_grid, 256, 0, stream>>>(zbuf, stats,
                                            bn_g + (size_t)l * NH,
                                            bn_b + (size_t)l * NH, hl, hn,
                                            N_NODES);
  }

  // JK: out = sum_l (pool(h_l) @ jk_W[l] + jk_b[l])
  hipMemsetAsync(out, 0, (size_t)N_GRAPH * NCLASS * sizeof(float), stream);
  for (int l = 0; l <= NLAYER; ++l) {
    hipMemsetAsync(pooled, 0, (size_t)N_GRAPH * NH * sizeof(float), stream);
    pool_add<<<pool_grid, 256, 0, stream>>>(hbuf + (size_t)l * HN, batch,
                                            pooled);
    jk_head<<<head_grid, 256, 0, stream>>>(pooled,
                                           jk_W + (size_t)l * NH * NCLASS,
                                           jk_b + (size_t)l * NCLASS, out);
  }
}